// WMVLoss_5600637354239
// MI455X (gfx1250) — compile-verified
//
#include <hip/hip_runtime.h>
#include <math.h>

// ---------------------------------------------------------------------------
// WMV pairwise margin loss on MI455X (gfx1250, wave32).
//
//   loss = (1/N) * sum_{i in neg, j in pos} max(gamma - (s_j - s_i), 0)^2
//   s = sigmoid(x1 - x0)   (softmax over 2 logits, class-1 prob)
//
// Bilinear core per 16x16 tile via V_WMMA_F32_16X16X4_F32:
//   m[i][j] = (neg_i*s_i)*pos_j + neg_i*(pos_j*(gamma-s_j))
// A(16x4): row i = [neg*s, neg, 0, 0]   (lanes 0-15 = rows M, lanes 16-31 = K2,3 -> 0)
// B(4x16): col j = [pos, pos*(g-s),0,0] (lanes 0-15 = K0/K1 rows, lanes 16-31 -> 0)
// Zero-padding of lanes 16-31 via a hoisted 0/1 mask float (no EXEC divergence;
// WMMA requires EXEC all-1s).
//
// Clamp-square without v_max canonicalize overhead:
//   u = m + |m| = 2*max(m,0)   (single v_add_f32 with |.| src modifier)
//   sum max(m,0)^2 = 0.25 * sum u^2  (0.25 folded into the wave partial once)
// ---------------------------------------------------------------------------

typedef __attribute__((ext_vector_type(2))) float v2f;
typedef __attribute__((ext_vector_type(8))) float v8f;

#define N_TOT 16384
#define GAMMA_F 0.3f
#define ITILES (N_TOT / 16)                // 1024 i-tiles
#define JCHUNKS 8                          // split j range for occupancy
#define TILES_PER_CHUNK (ITILES / JCHUNKS) // 128 j-tiles per chunk
#define WAVES_PER_BLOCK 8                  // 256 threads = 8 wave32
#define NPARTIALS (ITILES * JCHUNKS)       // 8192 wave partials

// ---------------- kernel 1: scores + positive flags ----------------
__global__ void wmv_score_kernel(const float* __restrict__ in,
                                 const long long* __restrict__ tgt,
                                 float2* __restrict__ SP) {
    int n = blockIdx.x * blockDim.x + threadIdx.x;
    if (n < N_TOT) {
        float x0 = in[2 * n + 0];
        float x1 = in[2 * n + 1];
        float s = 1.0f / (1.0f + __expf(x0 - x1));   // softmax(...)[:,1]
        float p = (tgt[n] == 1) ? 1.0f : 0.0f;
        SP[n] = make_float2(s, p);
    }
}

// ---------------- kernel 2: WMMA pairwise tiles ----------------
__global__ void wmv_pair_kernel(const float2* __restrict__ SP,
                                float* __restrict__ partials) {
    const int wave = blockIdx.x * WAVES_PER_BLOCK + (threadIdx.x >> 5); // i-tile id
    const int lane = threadIdx.x & 31;
    const int col  = lane & 15;                    // matrix column / row index
    const float lmask = (lane < 16) ? 1.0f : 0.0f; // 0 for K=2,3 padding lanes
    const int i0 = wave * 16;
    const int jc = blockIdx.y;
    const int j0 = jc * TILES_PER_CHUNK * 16;

    // A matrix (built once): lanes 0-15 hold rows M=0..15 (K=0 in .x, K=1 in .y);
    // lanes 16-31 hold K=2,3 which the mask zeroes.
    v2f a;
    {
        float2 sp = SP[i0 + col];                // lanes 16-31 reload lanes 0-15's data
        float neg = (1.0f - sp.y) * lmask;       // neg_i, zeroed on padding lanes
        a.x = neg * sp.x;                        // K=0: neg_i * s_i
        a.y = neg;                               // K=1: neg_i
    }

    float acc = 0.0f;
    for (int t = 0; t < TILES_PER_CHUNK; ++t) {
        // B matrix: lanes 0-15 hold rows K=0 (.x) / K=1 (.y), cols N=lane;
        // lanes 16-31 (K=2,3) zeroed via mask — no EXEC divergence in the loop.
        float2 sp = SP[j0 + t * 16 + col];
        float ly = sp.y * lmask;                 // pos_j, zeroed on padding lanes
        v2f b;
        b.x = ly;                                // K=0: pos_j
        b.y = ly * (GAMMA_F - sp.x);             // K=1: pos_j * (gamma - s_j)

        v8f c = {};
        c = __builtin_amdgcn_wmma_f32_16x16x4_f32(
            /*neg_a=*/false, a, /*neg_b=*/false, b,
            /*c_mod=*/(short)0, c, /*reuse_a=*/false, /*reuse_b=*/false);
#pragma unroll
        for (int v = 0; v < 8; ++v) {
            float u = c[v] + fabsf(c[v]);        // 2*max(m,0): v_add_f32 v, v, |v|
            acc = fmaf(u, u, acc);               // accumulates 4*max(m,0)^2
        }
    }
    acc *= 0.25f;                                // undo the factor of 2^2 once

    // Deterministic intra-wave tree reduction (wave32).
#pragma unroll
    for (int off = 16; off > 0; off >>= 1)
        acc += __shfl_xor(acc, off, 32);
    if (lane == 0)
        partials[jc * ITILES + wave] = acc;
}

// ---------------- kernel 3: deterministic final reduction ----------------
__global__ void wmv_reduce_kernel(const float* __restrict__ partials,
                                  float* __restrict__ out) {
    __shared__ float sm[256];
    float acc = 0.0f;
    for (int i = threadIdx.x; i < NPARTIALS; i += 256)
        acc += partials[i];
    sm[threadIdx.x] = acc;
    __syncthreads();
    for (int off = 128; off > 0; off >>= 1) {
        if (threadIdx.x < (unsigned)off) sm[threadIdx.x] += sm[threadIdx.x + off];
        __syncthreads();
    }
    if (threadIdx.x == 0)
        out[0] = sm[0] / (float)N_TOT;
}

extern "C" void kernel_launch(void* const* d_in, const int* in_sizes, int n_in,
                              void* d_out, int out_size, void* d_ws, size_t ws_size,
                              hipStream_t stream) {
    (void)in_sizes; (void)n_in; (void)out_size; (void)ws_size;
    const float* inputs = (const float*)d_in[0];          // [N, 2] f32
    const long long* target = (const long long*)d_in[1];  // [N] i64
    float* out = (float*)d_out;                           // scalar f32

    // Workspace layout: SP float2[N_TOT] (128 KB) | partials float[NPARTIALS] (32 KB)
    float2* SP = (float2*)d_ws;
    float* partials = (float*)((char*)d_ws + (size_t)N_TOT * sizeof(float2));

    wmv_score_kernel<<<dim3(N_TOT / 256), dim3(256), 0, stream>>>(inputs, target, SP);

    dim3 grid(ITILES / WAVES_PER_BLOCK, JCHUNKS);  // 128 x 8 blocks, 8 waves each
    wmv_pair_kernel<<<grid, dim3(256), 0, stream>>>(SP, partials);

    wmv_reduce_kernel<<<dim3(1), dim3(256), 0, stream>>>(partials, out);
}